// yoloLoss_89266600280302
// MI455X (gfx1250) — compile-verified
//
#include <hip/hip_runtime.h>
#include <math.h>

// Problem constants (fixed by setup_inputs)
#define BB 512
#define HH 28
#define WW 28
#define TT 30          // 5*BOX_NUM + C
#define KK 32          // objects per image == wave32 width
#define CC 20
#define HWCELLS (HH * WW)        // 784
#define NELEM (BB * HWCELLS * TT) // 12,042,240 floats (48.2 MB)

#define NOBJ_BLOCKS 1024
#define NOBJ_THREADS 256
#define NPARTIAL (NOBJ_BLOCKS + BB)   // 1536 = 24 * 64 -> exact WMMA tiling

typedef __attribute__((ext_vector_type(2))) float v2f;
typedef __attribute__((ext_vector_type(8))) float v8f;

// ---------------------------------------------------------------------------
// Kernel B: stream the full 48 MB tensor coalesced (float4), accumulate
// 0.5 * conf^2 for channels 4 and 9 of every 30-float cell record.
// Deterministic: fixed grid-stride order per thread + fixed LDS tree.
// ---------------------------------------------------------------------------
__global__ void noobj_partial_kernel(const float4* __restrict__ out4,
                                     float* __restrict__ partials, int n4) {
    __shared__ float red[NOBJ_THREADS];
    const int tid = threadIdx.x;
    int i = blockIdx.x * NOBJ_THREADS + tid;
    const int stride = NOBJ_BLOCKS * NOBJ_THREADS;
    float acc = 0.0f;
    for (; i < n4; i += stride) {
        const float4 v = out4[i];
        const int m = (4 * i) % TT;   // channel of v.x within its cell record
        if (m + 0 == 4 || m + 0 == 9) acc += v.x * v.x;
        if (m + 1 == 4 || m + 1 == 9) acc += v.y * v.y;
        if (m + 2 == 4 || m + 2 == 9) acc += v.z * v.z;
        if (m + 3 == 4 || m + 3 == 9) acc += v.w * v.w;
    }
    red[tid] = acc;
    __syncthreads();
    for (int off = NOBJ_THREADS / 2; off > 0; off >>= 1) {
        if (tid < off) red[tid] += red[tid + off];
        __syncthreads();
    }
    if (tid == 0) partials[blockIdx.x] = 0.5f * red[0];   // NOOBJ_RATE
}

// ---------------------------------------------------------------------------
// Kernel A: one wave32 per batch image, lane k = object k (K == 32).
// Computes IOUs, per-cell conflict resolution via 32 shuffles, and all
// sparse loss terms (center, size, conf-obj, class, noobj correction).
// ---------------------------------------------------------------------------
__global__ void winner_loss_kernel(const float* __restrict__ output,
                                   const float* __restrict__ truths,
                                   float* __restrict__ partials) {
    const int b = blockIdx.x;
    const int k = threadIdx.x;   // 0..31

    const float* t = truths + ((size_t)b * KK + k) * 5;
    const float x0 = t[0], y0 = t[1], x1 = t[2], y1 = t[3], clsf = t[4];

    const float tcx = 0.5f * (x0 + x1), tcy = 0.5f * (y0 + y1);
    const float tw = x1 - x0, th = y1 - y0;
    const float dx = tcx * (float)WW, dy = tcy * (float)HH;
    const float gxk = ceilf(dx) - 1.0f, gyk = ceilf(dy) - 1.0f;
    const float ddx = dx - gxk, ddy = dy - gyk;          // uses UNclipped gxk/gyk
    float posf = gyk * (float)WW + gxk;
    posf = fminf(fmaxf(posf, 0.0f), (float)(HWCELLS - 1));
    const int pos = (int)posf;

    const float* p = output + ((size_t)b * HWCELLS + pos) * TT;
    float pt[10];
#pragma unroll
    for (int i = 0; i < 10; ++i) pt[i] = p[i];

    const float col = (float)(pos % WW);
    const float row = (float)(pos / WW);

    float iou[2];
#pragma unroll
    for (int n = 0; n < 2; ++n) {
        const float px = pt[5 * n + 0], py = pt[5 * n + 1];
        const float pw = pt[5 * n + 2], ph = pt[5 * n + 3];
        const float pcx = (px + col) / (float)WW;
        const float pcy = (py + row) / (float)HH;
        const float pb0 = pcx - 0.5f * pw, pb1 = pcy - 0.5f * ph;
        const float pb2 = pcx + 0.5f * pw, pb3 = pcy + 0.5f * ph;
        const float ix0 = fmaxf(x0, pb0), iy0 = fmaxf(y0, pb1);
        const float ix1 = fminf(x1, pb2), iy1 = fminf(y1, pb3);
        const float inter = fmaxf(ix1 - ix0, 0.0f) * fmaxf(iy1 - iy0, 0.0f);
        const float uni = tw * th + (pb2 - pb0) * (pb3 - pb1) - inter;
        iou[n] = inter / fmaxf(uni, 1e-12f);
    }
    const int   ind  = (iou[1] > iou[0]) ? 1 : 0;        // argmax, ties -> box 0
    const float miou = fmaxf(iou[0], iou[1]);
    const int   valid = (((x0 + y0 + x1 + y1) != 0.0f) && (miou > 0.0f)) ? 1 : 0;

    // Per-cell conflict resolution: earliest object with max IOU wins the cell.
    int lose = 0;
    for (int j = 0; j < 32; ++j) {
        const int   posj = __shfl(pos, j, 32);
        const float iouj = __shfl(miou, j, 32);
        const int   vj   = __shfl(valid, j, 32);
        if (j != k && posj == pos && vj &&
            ((j < k && iouj >= miou) || (j > k && iouj > miou)))
            lose = 1;
    }

    float part = 0.0f;
    if (valid && !lose) {
        const float pw = pt[5 * ind + 2], ph = pt[5 * ind + 3], pc = pt[5 * ind + 4];
        const float cdx = pt[5 * ind + 0] - ddx;
        const float cdy = pt[5 * ind + 1] - ddy;
        part += 5.0f * (cdx * cdx + cdy * cdy);                       // COORD_RATE
        const float sw = sqrtf(pw) - sqrtf(tw);
        const float sh = sqrtf(ph) - sqrtf(th);
        part += 5.0f * (sw * sw + sh * sh);
        const float dc = pc - miou;
        part += dc * dc;                 // conf-obj
        part -= 0.5f * pc * pc;          // remove winning box from noobj sum
        const int c = (int)clsf;
#pragma unroll
        for (int cc = 0; cc < CC; ++cc) {
            const float q = p[10 + cc] - (cc == c ? 1.0f : 0.0f);
            part += q * q;               // class loss (mask = all classes)
        }
    }

    // wave32 tree reduction
    for (int off = 16; off > 0; off >>= 1) part += __shfl_down(part, off, 32);
    if (k == 0) partials[b] = part;
}

// ---------------------------------------------------------------------------
// Kernel C: final deterministic reduction of the 1536 partials using
// V_WMMA_F32_16X16X4_F32 chained accumulation: D = A x ones(4x16) + C.
// Each WMMA folds a 64-float tile into 16 row-sums (fp32 throughout); the
// all-ones B matrix makes the A-tile lane layout irrelevant to the total.
// One wave (EXEC all-1s as WMMA requires), no divergence before the loop.
// ---------------------------------------------------------------------------
__global__ void final_reduce_kernel(const float* __restrict__ partials,
                                    float* __restrict__ out) {
    const int lane = threadIdx.x;   // 0..31, full wave
    v8f c = {};
    v2f ones;
    ones.x = 1.0f;
    ones.y = 1.0f;
#pragma unroll
    for (int tile = 0; tile < NPARTIAL / 64; ++tile) {   // 24 tiles
        v2f a;
        a.x = partials[tile * 64 + lane];
        a.y = partials[tile * 64 + 32 + lane];
        c = __builtin_amdgcn_wmma_f32_16x16x4_f32(
                /*neg_a=*/false, a, /*neg_b=*/false, ones,
                /*c_mod=*/(short)0, c, /*reuse_a=*/false, /*reuse_b=*/false);
    }
    // C/D layout: lane L holds column L%16, rows r + 8*(L/16) in c[r].
    // Column 0 total = (rows 0-7 in lane 0) + (rows 8-15 in lane 16).
    float local = c[0] + c[1] + c[2] + c[3] + c[4] + c[5] + c[6] + c[7];
    const float other = __shfl(local, lane ^ 16, 32);
    const float tot = local + other;
    if (lane == 0) out[0] = tot / (float)BB;
}

// ---------------------------------------------------------------------------
extern "C" void kernel_launch(void* const* d_in, const int* in_sizes, int n_in,
                              void* d_out, int out_size, void* d_ws, size_t ws_size,
                              hipStream_t stream) {
    const float* output = (const float*)d_in[0];  // (512,28,28,30) fp32
    const float* truths = (const float*)d_in[1];  // (512,32,5)    fp32
    (void)in_sizes; (void)n_in; (void)out_size; (void)ws_size;

    float* ws = (float*)d_ws;   // [0,1024): noobj partials, [1024,1536): image partials

    const int n4 = NELEM / 4;   // 3,010,560 float4 elements
    noobj_partial_kernel<<<NOBJ_BLOCKS, NOBJ_THREADS, 0, stream>>>(
        (const float4*)output, ws, n4);
    winner_loss_kernel<<<BB, 32, 0, stream>>>(output, truths, ws + NOBJ_BLOCKS);
    final_reduce_kernel<<<1, 32, 0, stream>>>(ws, (float*)d_out);
}